// Encoder1_26405458936002
// MI455X (gfx1250) — compile-verified
//
#include <hip/hip_runtime.h>

#define N_NODES 100000
#define N_EDGES 3200000
#define D 128
#define BN_EPS 1e-5f

typedef __attribute__((ext_vector_type(2))) float v2f;
typedef __attribute__((ext_vector_type(8))) float v8f;

// ---------------------------------------------------------------- utilities
__global__ void zero_f32(float* __restrict__ p, int n) {
    int i = blockIdx.x * blockDim.x + threadIdx.x;
    if (i < n) p[i] = 0.0f;
}

// ------------------------------------------------------------- degree pass
__global__ void edge_degrees(const int* __restrict__ src, const int* __restrict__ dst,
                             float* __restrict__ degout, float* __restrict__ degin, int ne) {
    int i = blockIdx.x * blockDim.x + threadIdx.x;
    if (i < ne) {
        unsafeAtomicAdd(&degout[src[i]], 1.0f);
        unsafeAtomicAdd(&degin[dst[i]], 1.0f);
    }
}

__global__ void deg_norm(const float* __restrict__ degout, const float* __restrict__ degin,
                         float* __restrict__ nsrc, float* __restrict__ ndst, int n) {
    int i = blockIdx.x * blockDim.x + threadIdx.x;
    if (i < n) {
        nsrc[i] = rsqrtf(fmaxf(degout[i], 1.0f));
        ndst[i] = rsqrtf(fmaxf(degin[i], 1.0f));
    }
}

// --------------------------------------------------------------- WMMA GEMM
// HW = diag(nsrc) * H * W ; H is [nrows x 128], W is [128 x 128].
// One block = 16-row stripe; 8 waves, wave w computes cols [16w, 16w+16).
// f32 WMMA 16x16x4: A lane layout: lanes 0-15 -> {K, K+1}, lanes 16-31 -> {K+2, K+3},
// row M = lane&15. B mirrors it with N = lane&15. C: vgpr v -> row v + 8*(lane>=16),
// col = lane&15.
__global__ void gcn_gemm_wmma(const float* __restrict__ H, const float* __restrict__ nsrc,
                              const float* __restrict__ W, float* __restrict__ HW, int nrows) {
    const int lane = threadIdx.x & 31;
    const int wave = threadIdx.x >> 5;          // 0..7 -> column tile
    const int row0 = blockIdx.x * 16;           // nrows % 16 == 0 (100000 = 6250*16)
    const int col0 = wave * 16;

    const int m    = lane & 15;
    const int koff = (lane >> 4) * 2;           // 0 or 2
    const int row  = row0 + m;
    const float s  = nsrc[row];
    const float* __restrict__ hrow = H + (size_t)row * D;
    const int n = col0 + (lane & 15);

    v8f acc = {};
    #pragma unroll 4
    for (int k = 0; k < D; k += 4) {
        v2f a, b;
        a.x = hrow[k + koff]     * s;
        a.y = hrow[k + koff + 1] * s;
        b.x = W[(size_t)(k + koff) * D + n];
        b.y = W[(size_t)(k + koff + 1) * D + n];
        acc = __builtin_amdgcn_wmma_f32_16x16x4_f32(
            /*neg_a=*/false, a, /*neg_b=*/false, b,
            /*c_mod=*/(short)0, acc, /*reuse_a=*/false, /*reuse_b=*/false);
    }

    const int crow0 = row0 + ((lane >> 4) * 8);
    const int col   = col0 + (lane & 15);
    #pragma unroll
    for (int v = 0; v < 8; ++v)
        HW[(size_t)(crow0 + v) * D + col] = acc[v];
}

// ----------------------------------------------------- edge gather/scatter
// One wave per edge: 32 lanes x float4 = 128 floats. Gather hw[src], atomically
// add into agg[dst]. hw/agg are 51.2 MB each -> resident in the 192 MB L2.
__global__ void edge_scatter(const int* __restrict__ src, const int* __restrict__ dst,
                             const float* __restrict__ HW, float* __restrict__ AGG, int ne) {
    int wid  = (int)((blockIdx.x * (unsigned)blockDim.x + threadIdx.x) >> 5);
    int lane = threadIdx.x & 31;
    if (wid >= ne) return;
    int s = src[wid];
    int d = dst[wid];
    const float4* __restrict__ p = (const float4*)(HW + (size_t)s * D);
    float4 v = p[lane];
    float* q = AGG + (size_t)d * D + lane * 4;
    unsafeAtomicAdd(q + 0, v.x);
    unsafeAtomicAdd(q + 1, v.y);
    unsafeAtomicAdd(q + 2, v.z);
    unsafeAtomicAdd(q + 3, v.w);
}

// ----------------------------------------------------------- BN statistics
// Stats of y = agg * ndst (conv bias cancels inside BN). 256 threads = 2 nodes
// per step, channel = tid & 127. LDS-combine then one atomic per channel/block.
__global__ void bn_stats(const float* __restrict__ AGG, const float* __restrict__ ndst,
                         float* __restrict__ sums, float* __restrict__ sumsq, int nnodes) {
    __shared__ float sh1[256];
    __shared__ float sh2[256];
    const int c   = threadIdx.x & 127;
    const int sub = threadIdx.x >> 7;
    float s = 0.0f, s2 = 0.0f;
    for (int node = blockIdx.x * 2 + sub; node < nnodes; node += gridDim.x * 2) {
        float y = AGG[(size_t)node * D + c] * ndst[node];
        s += y;
        s2 += y * y;
    }
    sh1[threadIdx.x] = s;
    sh2[threadIdx.x] = s2;
    __syncthreads();
    if (threadIdx.x < 128) {
        unsafeAtomicAdd(&sums[c],  sh1[threadIdx.x] + sh1[threadIdx.x + 128]);
        unsafeAtomicAdd(&sumsq[c], sh2[threadIdx.x] + sh2[threadIdx.x + 128]);
    }
}

__global__ void bn_finalize(const float* __restrict__ sums, const float* __restrict__ sumsq,
                            const float* __restrict__ g, const float* __restrict__ be,
                            float* __restrict__ scale, float* __restrict__ shift, int nnodes) {
    int c = threadIdx.x;                         // 128 threads
    float invn = 1.0f / (float)nnodes;
    float mu   = sums[c] * invn;
    float var  = sumsq[c] * invn - mu * mu;
    float rstd = rsqrtf(var + BN_EPS);
    float sc   = rstd * g[c];
    scale[c] = sc;
    shift[c] = be[c] - mu * sc;
}

// ------------------------------------------------------ fused BN + PReLU
__global__ void bn_apply(const float* __restrict__ AGG, const float* __restrict__ ndst,
                         const float* __restrict__ scale, const float* __restrict__ shift,
                         const float* __restrict__ alpha, float* __restrict__ out, int total) {
    int i = blockIdx.x * blockDim.x + threadIdx.x;
    if (i >= total) return;
    int c    = i & (D - 1);
    int node = i >> 7;
    float y = AGG[i] * ndst[node];
    float t = y * scale[c] + shift[c];
    float a = alpha[0];
    out[i] = (t >= 0.0f) ? t : a * t;
}

// ---------------------------------------------------------------- launcher
extern "C" void kernel_launch(void* const* d_in, const int* in_sizes, int n_in,
                              void* d_out, int out_size, void* d_ws, size_t ws_size,
                              hipStream_t stream) {
    (void)in_sizes; (void)n_in; (void)out_size; (void)ws_size;

    const float* feat = (const float*)d_in[0];
    const int*   ei   = (const int*)d_in[1];
    const int*   src  = ei;
    const int*   dst  = ei + N_EDGES;
    const float* W1   = (const float*)d_in[2];
    const float* g1   = (const float*)d_in[4];
    const float* be1  = (const float*)d_in[5];
    const float* a1   = (const float*)d_in[6];
    const float* W2   = (const float*)d_in[7];
    const float* g2   = (const float*)d_in[9];
    const float* be2  = (const float*)d_in[10];
    const float* a2   = (const float*)d_in[11];

    const int NF = N_NODES * D;                  // 12,800,000

    float* ws     = (float*)d_ws;
    float* hw     = ws;                          // [N_NODES*D]
    float* agg    = ws + (size_t)NF;             // [N_NODES*D]
    float* h1     = ws + (size_t)2 * NF;         // [N_NODES*D]
    float* degout = ws + (size_t)3 * NF;         // [N_NODES]
    float* degin  = degout + N_NODES;
    float* nsrc   = degin  + N_NODES;
    float* ndst   = nsrc   + N_NODES;
    float* sums   = ndst   + N_NODES;            // [128]
    float* sumsq  = sums   + D;                  // [128]
    float* scale  = sumsq  + D;                  // [128]
    float* shift  = scale  + D;                  // [128]

    const int B = 256;
    dim3 blk(B);

    // degrees + normalization factors (shared by both layers)
    zero_f32<<<dim3((2 * N_NODES + B - 1) / B), blk, 0, stream>>>(degout, 2 * N_NODES);
    edge_degrees<<<dim3((N_EDGES + B - 1) / B), blk, 0, stream>>>(src, dst, degout, degin, N_EDGES);
    deg_norm<<<dim3((N_NODES + B - 1) / B), blk, 0, stream>>>(degout, degin, nsrc, ndst, N_NODES);

    const dim3 gemm_grid(N_NODES / 16);          // 6250 (exact)
    const dim3 scat_grid((N_EDGES * 32u + B - 1) / B);
    const dim3 elem_grid((NF + B - 1) / B);
    const dim3 stat_grid(512);

    // ---------------- layer 1 ----------------
    gcn_gemm_wmma<<<gemm_grid, blk, 0, stream>>>(feat, nsrc, W1, hw, N_NODES);
    zero_f32<<<elem_grid, blk, 0, stream>>>(agg, NF);
    zero_f32<<<dim3(1), blk, 0, stream>>>(sums, 2 * D);
    edge_scatter<<<scat_grid, blk, 0, stream>>>(src, dst, hw, agg, N_EDGES);
    bn_stats<<<stat_grid, blk, 0, stream>>>(agg, ndst, sums, sumsq, N_NODES);
    bn_finalize<<<dim3(1), dim3(D), 0, stream>>>(sums, sumsq, g1, be1, scale, shift, N_NODES);
    bn_apply<<<elem_grid, blk, 0, stream>>>(agg, ndst, scale, shift, a1, h1, NF);

    // ---------------- layer 2 ----------------
    gcn_gemm_wmma<<<gemm_grid, blk, 0, stream>>>(h1, nsrc, W2, hw, N_NODES);
    zero_f32<<<elem_grid, blk, 0, stream>>>(agg, NF);
    zero_f32<<<dim3(1), blk, 0, stream>>>(sums, 2 * D);
    edge_scatter<<<scat_grid, blk, 0, stream>>>(src, dst, hw, agg, N_EDGES);
    bn_stats<<<stat_grid, blk, 0, stream>>>(agg, ndst, sums, sumsq, N_NODES);
    bn_finalize<<<dim3(1), dim3(D), 0, stream>>>(sums, sumsq, g2, be2, scale, shift, N_NODES);
    bn_apply<<<elem_grid, blk, 0, stream>>>(agg, ndst, scale, shift, a2, (float*)d_out, NF);
}